// A3Net_36618891165870
// MI455X (gfx1250) — compile-verified
//
#include <hip/hip_runtime.h>

// ---------------------------------------------------------------------------
// Spiking NN (snntorch Leaky, reset-by-subtraction), T=20, B=262144, 3->24->4.
//
// Roofline: 147 MB mandatory traffic @ 23.3 TB/s ~= 6.3 us; ~1.76 GFLOP total
// -> memory/VALU bound. Time recurrence forces per-wave ownership of batch
// rows across all 20 steps with membrane potentials in registers.
//
// Matrix ops: V_WMMA_F32_16X16X4_F32 (the only f32 WMMA shape on CDNA5).
//  layer1: D1 = W1 (24x3 -> two 16x4 A tiles, K padded 3->4) @ x^T (3x16 B)
//  layer2: D2 = W2 (4x24 -> eight 16x4 A tiles, K-permuted)  @ spk1^T
// Batch is mapped to N in BOTH layers, so mem1/mem2/spikes stay in the C/D
// accumulator layout (lane = batch col, VGPR = hidden/class row) and the LIF
// decay+reset is folded into the WMMA C operand.
//
// K-permutation trick (kills all cross-lane shuffles): the contraction dim of
// layer 2 may be reordered freely if A and B agree. Spike element (h, n) sits
// at vgpr h%8, lower lanes for h%16<8, upper lanes for h%16>=8. Choosing the
// K=4 quad (2s, 2s+1, 2s+8, 2s+9) makes the B operand EXACTLY the register
// pair (sp[2s], sp[2s+1]) as produced by layer 1 — the W2 A tiles are built
// once with the matching column permutation. Hidden 16..23 use quads
// (16+2u, 17+2u, pad, pad) with zeroed A columns. 8 WMMAs replace
// 6 WMMAs + 12 ds_bpermute + 12 cndmask + s_wait_dscnt stalls.
//
// Layout assumptions (CDNA5 ISA, f32 16x16x4):
//  A 16x4:  lanes 0-15 = rows M, vgpr0/1 = K0/K1; lanes 16-31 = K2/K3.
//  B 4x16:  mirror of A: vgpr0 = {K0 | K2}, vgpr1 = {K1 | K3}, N = lane%16.
//  C/D 16x16: vgpr v, lanes 0-15 -> (M=v, N=lane); lanes 16-31 -> (M=v+8).
// ---------------------------------------------------------------------------

#define THRESH 1.0f
#define BETA   0.9f

constexpr int T  = 20;
constexpr int Bn = 262144;
constexpr int F  = 3;
constexpr int H  = 24;
constexpr int C  = 4;

typedef float v2f __attribute__((ext_vector_type(2)));
typedef float v4f __attribute__((ext_vector_type(4)));
typedef float v8f __attribute__((ext_vector_type(8)));

static __device__ inline v8f wmma_f32(v2f a, v2f b, v8f c) {
  // 8-arg form: (neg_a, A, neg_b, B, c_mod, C, reuse_a, reuse_b)
  return __builtin_amdgcn_wmma_f32_16x16x4_f32(false, a, false, b, (short)0, c,
                                               false, false);
}

static __device__ inline float leak_reset(float m) {
  // beta*mem - reset*THRESHOLD, reset = Heaviside(mem - THRESHOLD)
  return BETA * m - ((m > THRESH) ? THRESH : 0.0f);
}
static __device__ inline float spike(float m) {
  return (m > THRESH) ? 1.0f : 0.0f;
}

__global__ void __launch_bounds__(256) snn_lif_wmma(
    const float* __restrict__ x,   // (T, B, F)
    const float* __restrict__ W1,  // (H, F)
    const float* __restrict__ W2,  // (C, H)
    float* __restrict__ out)       // (T, B, C) spikes
{
  const int lane = threadIdx.x & 31;
  const int lo   = lane & 15;
  const bool hi  = lane >= 16;
  const int wave = threadIdx.x >> 5;
  const int b0   = (blockIdx.x * (blockDim.x >> 5) + wave) * 16;
  const int b    = b0 + lo;  // this lane's batch column (N index)

  // ---- A tiles for W1: rows 16g..16g+15, K = F padded to 4 ----
  v2f aW1[2];
#pragma unroll
  for (int g = 0; g < 2; ++g) {
    const int m = 16 * g + lo;
    const int k = hi ? 2 : 0;                          // K0/K2 slots
    aW1[g].x = (m < H) ? W1[m * F + k] : 0.0f;         // k in {0,2} < F
    aW1[g].y = (!hi && m < H) ? W1[m * F + 1] : 0.0f;  // K1; K3 is zero pad
  }

  // ---- A tiles for W2, K-permuted to match resident spike layout ----
  // chunks 0..3: K quad (2s, 2s+1, 2s+8, 2s+9)        (hidden 0..15)
  // chunks 4..7: K quad (16+2u, 17+2u, pad, pad)      (hidden 16..23)
  v2f aW2[8];
#pragma unroll
  for (int s = 0; s < 4; ++s) {
    aW2[s].x = (lo < C) ? W2[lo * H + 2 * s + (hi ? 8 : 0)] : 0.0f;
    aW2[s].y = (lo < C) ? W2[lo * H + 2 * s + (hi ? 9 : 1)] : 0.0f;
  }
#pragma unroll
  for (int u = 0; u < 4; ++u) {
    aW2[4 + u].x = (lo < C && !hi) ? W2[lo * H + 16 + 2 * u] : 0.0f;
    aW2[4 + u].y = (lo < C && !hi) ? W2[lo * H + 17 + 2 * u] : 0.0f;
  }

  // Membrane state, resident in WMMA C/D layout for the whole run.
  v8f mem1[2] = {};  // tile g: vgpr v <-> hidden 16g+v (+8 upper half)
  v8f mem2    = {};  // vgpr v <-> class v (0..3 valid); vgprs 4..7 stay 0

  for (int t = 0; t < T; ++t) {
    // ---- B tile of x^T (K=F x N=16): vgpr0 = {f0 | f2}, vgpr1 = {f1 | 0} ----
    const size_t base = ((size_t)t * Bn + b) * F;
    v2f bx;
    bx.x = __builtin_nontemporal_load(&x[base + (hi ? 2u : 0u)]);
    bx.y = hi ? 0.0f : __builtin_nontemporal_load(&x[base + 1]);
    if (t + 1 < T)  // next timestep slab is 3 MB away; help the prefetcher
      __builtin_prefetch(&x[base + (size_t)Bn * F], 0, 1);

    // ---- layer 1: mem1 = cur + (beta*mem1 - reset), reset folded into C ----
#pragma unroll
    for (int g = 0; g < 2; ++g) {
      v8f cin;
#pragma unroll
      for (int i = 0; i < 8; ++i) cin[i] = leak_reset(mem1[g][i]);
      mem1[g] = wmma_f32(aW1[g], bx, cin);
    }

    // ---- layer-1 spikes (D layout; these ARE the layer-2 B operands) ----
    v8f sp0, sp1;
#pragma unroll
    for (int i = 0; i < 8; ++i) {
      sp0[i] = spike(mem1[0][i]);
      sp1[i] = spike(mem1[1][i]);
    }

    // ---- layer 2: eight chained K=4 WMMAs over the permuted K quads ----
    // mem2 pad vgprs (classes >= 4) are identically zero: leak only 0..3.
    v8f c2;
#pragma unroll
    for (int i = 0; i < 4; ++i) c2[i] = leak_reset(mem2[i]);
#pragma unroll
    for (int i = 4; i < 8; ++i) c2[i] = 0.0f;

    v8f acc = wmma_f32(aW2[0], (v2f){sp0[0], sp0[1]}, c2);
    acc = wmma_f32(aW2[1], (v2f){sp0[2], sp0[3]}, acc);
    acc = wmma_f32(aW2[2], (v2f){sp0[4], sp0[5]}, acc);
    acc = wmma_f32(aW2[3], (v2f){sp0[6], sp0[7]}, acc);
    acc = wmma_f32(aW2[4], (v2f){sp1[0], sp1[1]}, acc);
    acc = wmma_f32(aW2[5], (v2f){sp1[2], sp1[3]}, acc);
    acc = wmma_f32(aW2[6], (v2f){sp1[4], sp1[5]}, acc);
    acc = wmma_f32(aW2[7], (v2f){sp1[6], sp1[7]}, acc);
    mem2 = acc;

    // ---- output spikes: class c = vgpr c, batch = lane (lanes 0..15) ----
    if (!hi) {
      v4f o;
      o.x = spike(mem2[0]);
      o.y = spike(mem2[1]);
      o.z = spike(mem2[2]);
      o.w = spike(mem2[3]);
      // streamed once, never re-read on device -> nontemporal b128 store
      __builtin_nontemporal_store(
          o, reinterpret_cast<v4f*>(&out[((size_t)t * Bn + b) * C]));
    }
  }
}

extern "C" void kernel_launch(void* const* d_in, const int* in_sizes, int n_in,
                              void* d_out, int out_size, void* d_ws,
                              size_t ws_size, hipStream_t stream) {
  (void)in_sizes; (void)n_in; (void)out_size; (void)d_ws; (void)ws_size;
  const float* x  = (const float*)d_in[0];
  const float* W1 = (const float*)d_in[1];
  const float* W2 = (const float*)d_in[2];
  float* out      = (float*)d_out;

  // 8 waves/block * 16 batch rows/wave = 128 batch rows per block.
  dim3 block(256);
  dim3 grid(Bn / 128);  // 2048 blocks -> 16384 waves, T-loop carried in regs
  snn_lif_wmma<<<grid, block, 0, stream>>>(x, W1, W2, out);
}